// _TripletMessageLight_56435870270130
// MI455X (gfx1250) — compile-verified
//
#include <hip/hip_runtime.h>
#include <math.h>

typedef __attribute__((ext_vector_type(2))) float v2f;
typedef __attribute__((ext_vector_type(8))) float v8f;

#define C_DIM 64
#define CE_DIM 32
#define NEG_SLOPE 0.2f

// ---- monotonic float <-> uint encoding for atomicMax on floats ----
__device__ __forceinline__ unsigned f2ord(float f) {
    unsigned u = __float_as_uint(f);
    return (u & 0x80000000u) ? ~u : (u | 0x80000000u);
}
__device__ __forceinline__ float ord2f(unsigned u) {
    return (u & 0x80000000u) ? __uint_as_float(u & 0x7FFFFFFFu)
                             : __uint_as_float(~u);
}

// ======================================================================
// Kernel 1: xw = x @ W  via V_WMMA_F32_16X16X4_F32.
// One wave per 16x16 output tile; K=64 looped in steps of 4.
// ======================================================================
__global__ void gemm_xw_wmma(const float* __restrict__ x,
                             const float* __restrict__ W,
                             float* __restrict__ xw,
                             int n_nodes, int total_waves) {
    const int wave = (int)((blockIdx.x * blockDim.x + threadIdx.x) >> 5);
    if (wave >= total_waves) return;            // wave-uniform guard
    const int lane = threadIdx.x & 31;
    const int tiles_n = C_DIM / 16;             // 4 column tiles
    const int tile_m = wave / tiles_n;
    const int tile_n = wave % tiles_n;

    const int half = lane >> 4;                 // 0: lanes 0-15, 1: lanes 16-31
    const int lid  = lane & 15;
    int arow = tile_m * 16 + lid;               // A row this lane feeds
    if (arow >= n_nodes) arow = n_nodes - 1;    // clamp (N=50000 is exact anyway)
    const int ncol = tile_n * 16 + lid;         // B/C/D column this lane holds

    v8f acc = {};
    #pragma unroll
    for (int kk = 0; kk < C_DIM; kk += 4) {
        const int ka = kk + 2 * half;           // this lane's K pair
        // A 16x4: VGPR0 -> K=ka, VGPR1 -> K=ka+1 (8B-aligned contiguous pair)
        v2f a = *(const v2f*)(x + (size_t)arow * C_DIM + ka);
        // B 4x16: same per-VGPR K striping, N striped across lanes
        v2f b;
        b.x = W[(size_t)ka       * C_DIM + ncol];
        b.y = W[(size_t)(ka + 1) * C_DIM + ncol];
        acc = __builtin_amdgcn_wmma_f32_16x16x4_f32(
            /*neg_a=*/false, a, /*neg_b=*/false, b,
            /*c_mod=*/(short)0, acc, /*reuse_a=*/false, /*reuse_b=*/false);
    }
    // D 16x16 f32: VGPR v holds row tile_m*16 + 8*half + v, col ncol
    #pragma unroll
    for (int v = 0; v < 8; ++v) {
        const int row = tile_m * 16 + 8 * half + v;
        if (row < n_nodes) xw[(size_t)row * C_DIM + ncol] = acc[v];
    }
}

// ======================================================================
// Kernel 2: init out = bias (broadcast), segmax = -inf (ordered), segsum = 0
// ======================================================================
__global__ void init_state(float* __restrict__ out,
                           const float* __restrict__ bias,
                           unsigned* __restrict__ segmax,
                           float* __restrict__ segsum,
                           int n_nodes) {
    const int tid = blockIdx.x * blockDim.x + threadIdx.x;
    const int total = n_nodes * C_DIM;
    if (tid < total) out[tid] = bias[tid & (C_DIM - 1)];
    if (tid < n_nodes) {
        segmax[tid] = 0x007FFFFFu;  // f2ord(-inf)
        segsum[tid] = 0.0f;
    }
}

// ======================================================================
// Kernel 3: wave per edge -> attention logit, leaky relu, atomicMax per dst
//   alpha = sum(x_i*wa[0:64]) + sum(ea*wa[64:96]) + sum(x_j*wa[96:160])
// ======================================================================
__global__ void edge_logit(const float* __restrict__ xw,
                           const int* __restrict__ ei,
                           const float* __restrict__ ea,
                           const float* __restrict__ wa,
                           float* __restrict__ alpha,
                           unsigned* __restrict__ segmax,
                           int n_edges) {
    const int wave = (int)((blockIdx.x * blockDim.x + threadIdx.x) >> 5);
    if (wave >= n_edges) return;
    const int lane = threadIdx.x & 31;
    const int src = ei[wave];               // row 0 = source j
    const int dst = ei[n_edges + wave];     // row 1 = target i
    const float* xi = xw + (size_t)dst * C_DIM;
    const float* xj = xw + (size_t)src * C_DIM;
    const float* e  = ea + (size_t)wave * CE_DIM;

    float s = xi[lane]      * wa[lane]
            + xi[lane + 32] * wa[lane + 32]
            + e[lane]       * wa[64 + lane]
            + xj[lane]      * wa[96 + lane]
            + xj[lane + 32] * wa[128 + lane];
    #pragma unroll
    for (int off = 16; off > 0; off >>= 1)
        s += __shfl_xor(s, off, 32);

    if (lane == 0) {
        const float a = s > 0.0f ? s : NEG_SLOPE * s;
        alpha[wave] = a;
        atomicMax(segmax + dst, f2ord(a));
    }
}

// ======================================================================
// Kernel 4: thread per edge -> e = exp(alpha - max[dst]); segsum[dst] += e
// ======================================================================
__global__ void edge_exp(const int* __restrict__ ei,
                         const unsigned* __restrict__ segmax,
                         float* __restrict__ alpha,
                         float* __restrict__ segsum,
                         int n_edges) {
    const int eidx = blockIdx.x * blockDim.x + threadIdx.x;
    if (eidx >= n_edges) return;
    const int dst = ei[n_edges + eidx];
    const float m = ord2f(segmax[dst]);
    const float v = expf(alpha[eidx] - m);
    alpha[eidx] = v;
    atomicAdd(segsum + dst, v);
}

// ======================================================================
// Kernel 5: wave per edge -> out[dst] += (e / (sum+eps)) * x_j
// ======================================================================
__global__ void edge_scatter(const float* __restrict__ xw,
                             const int* __restrict__ ei,
                             const float* __restrict__ alpha,
                             const float* __restrict__ segsum,
                             float* __restrict__ out,
                             int n_edges) {
    const int wave = (int)((blockIdx.x * blockDim.x + threadIdx.x) >> 5);
    if (wave >= n_edges) return;
    const int lane = threadIdx.x & 31;
    const int src = ei[wave];
    const int dst = ei[n_edges + wave];
    const float coef = alpha[wave] / (segsum[dst] + 1e-16f);
    const float* xj = xw + (size_t)src * C_DIM;
    float* o = out + (size_t)dst * C_DIM;
    atomicAdd(o + lane,      coef * xj[lane]);
    atomicAdd(o + lane + 32, coef * xj[lane + 32]);
}

// ======================================================================
extern "C" void kernel_launch(void* const* d_in, const int* in_sizes, int n_in,
                              void* d_out, int out_size, void* d_ws, size_t ws_size,
                              hipStream_t stream) {
    const float* x    = (const float*)d_in[0];   // [N, 64]
    const int*   ei   = (const int*)  d_in[1];   // [2, E]
    const float* ea   = (const float*)d_in[2];   // [E, 32]
    const float* W    = (const float*)d_in[3];   // [64, 64]
    const float* wa   = (const float*)d_in[4];   // [1, 160]
    const float* bias = (const float*)d_in[5];   // [64]
    float* out = (float*)d_out;                  // [N, 64]

    const int n_nodes = in_sizes[0] / C_DIM;
    const int n_edges = in_sizes[1] / 2;

    // workspace carve (fp32, naturally aligned)
    float*    xw     = (float*)d_ws;                          // N*64
    float*    alpha  = xw + (size_t)n_nodes * C_DIM;          // E
    unsigned* segmax = (unsigned*)(alpha + n_edges);          // N
    float*    segsum = (float*)(segmax + n_nodes);            // N

    const int threads = 256;

    // 1) GEMM via WMMA
    const int tiles_m = (n_nodes + 15) / 16;
    const int total_waves = tiles_m * (C_DIM / 16);
    const int gemm_blocks = (total_waves * 32 + threads - 1) / threads;
    gemm_xw_wmma<<<gemm_blocks, threads, 0, stream>>>(x, W, xw, n_nodes, total_waves);

    // 2) init out/segment state
    const int init_total = n_nodes * C_DIM;
    init_state<<<(init_total + threads - 1) / threads, threads, 0, stream>>>(
        out, bias, segmax, segsum, n_nodes);

    // 3) per-edge logits + segment max
    const long long ew_threads = (long long)n_edges * 32;
    const int ew_blocks = (int)((ew_threads + threads - 1) / threads);
    edge_logit<<<ew_blocks, threads, 0, stream>>>(xw, ei, ea, wa, alpha, segmax, n_edges);

    // 4) exp + segment sum
    edge_exp<<<(n_edges + threads - 1) / threads, threads, 0, stream>>>(
        ei, segmax, alpha, segsum, n_edges);

    // 5) normalize + weighted scatter
    edge_scatter<<<ew_blocks, threads, 0, stream>>>(xw, ei, alpha, segsum, out, n_edges);
}